// SoftAugmentationAttention_18605798326657
// MI455X (gfx1250) — compile-verified
//
#include <hip/hip_runtime.h>
#include <hip/hip_bf16.h>
#include <stdint.h>

typedef __attribute__((ext_vector_type(2))) float v2f;
typedef __attribute__((ext_vector_type(8))) float v8f;
typedef __attribute__((ext_vector_type(4))) int   v4i;

// ---------------------------------------------------------------------------
// Kernel 1: q_proj[M,N] = hidden[M,K] @ weight[K,N]   (M=8192, K=N=768) FP32.
// 32x32 output tile per wave: 2 A-frags x 2 B-frags -> 4 accumulators,
// 4 x V_WMMA_F32_16X16X4_F32 per K-step (8 FLOP/byte from cache).
// A frag (16x4): lanes 0-15 hold M=lane rows; per-lane-half K pairs ->
//                one contiguous 8B load per lane per frag.
// D (16x16 f32): VGPR v -> M = v + 8*(lane>=16), N = lane&15.
// ---------------------------------------------------------------------------
__global__ void __launch_bounds__(256)
qproj_wmma_f32(const float* __restrict__ A,   // [8192, 768] hidden
               const float* __restrict__ W,   // [768, 768]  weight
               float* __restrict__ Q)         // [8192, 768] q_proj
{
    constexpr int K = 768, N = 768;
    constexpr int tilesN = N / 32;            // 24

    const int wave = threadIdx.x >> 5;        // 0..7
    const int lane = threadIdx.x & 31;
    const int tile = blockIdx.x * 8 + wave;   // 768 WGs * 8 = 6144 tiles, exact
    const int tm   = tile / tilesN;           // 0..255
    const int tn   = tile - tm * tilesN;      // 0..23

    const int half = lane >> 4;               // 0 or 1
    const int l16  = lane & 15;

    const float* __restrict__ a0p = A + (size_t)(tm * 32 + l16) * K;
    const float* __restrict__ a1p = a0p + (size_t)16 * K;
    const float* __restrict__ b0p = W + tn * 32 + l16;
    const float* __restrict__ b1p = b0p + 16;

    v8f c00 = {}, c01 = {}, c10 = {}, c11 = {};
    #pragma unroll 2
    for (int k = 0; k < K; k += 4) {
        const int ka = k + half * 2;
        v2f a0, a1, b0, b1;
        a0.x = a0p[ka];              a0.y = a0p[ka + 1];
        a1.x = a1p[ka];              a1.y = a1p[ka + 1];
        b0.x = b0p[(size_t)ka * N];  b0.y = b0p[(size_t)(ka + 1) * N];
        b1.x = b1p[(size_t)ka * N];  b1.y = b1p[(size_t)(ka + 1) * N];
        c00 = __builtin_amdgcn_wmma_f32_16x16x4_f32(false, a0, false, b0, (short)0, c00, false, false);
        c01 = __builtin_amdgcn_wmma_f32_16x16x4_f32(false, a0, false, b1, (short)0, c01, false, false);
        c10 = __builtin_amdgcn_wmma_f32_16x16x4_f32(false, a1, false, b0, (short)0, c10, false, false);
        c11 = __builtin_amdgcn_wmma_f32_16x16x4_f32(false, a1, false, b1, (short)0, c11, false, false);
    }

    float* __restrict__ q00 = Q + (size_t)(tm * 32 + half * 8) * N + tn * 32 + l16;
    #pragma unroll
    for (int v = 0; v < 8; ++v) {
        q00[(size_t)v * N]             = c00[v];
        q00[(size_t)v * N + 16]        = c01[v];
        q00[(size_t)(v + 16) * N]      = c10[v];
        q00[(size_t)(v + 16) * N + 16] = c11[v];
    }
}

// ---------------------------------------------------------------------------
// Kernel 2: per (b,l) row -- async-gather 8 embedding rows into LDS (ASYNCcnt
// path, no VGPR round-trip), bilinear scores with wave32 reductions, softmax
// over K=8, weighted sum. One 256-thread (8-wave) WG per row.
// ---------------------------------------------------------------------------
#if defined(__AMDGCN__) && __has_builtin(__builtin_amdgcn_global_load_async_to_lds_b128)
#define USE_ASYNC_LDS 1
#else
#define USE_ASYNC_LDS 0
#endif

__global__ void __launch_bounds__(256)
attn_softmax_kernel(const float* __restrict__ Q,          // [8192, 768] q_proj
                    const long long* __restrict__ idx,    // [8192, 8] int64
                    const float* __restrict__ w2v,        // [50000, 768]
                    float* __restrict__ out)              // [8192, 768]
{
    constexpr int E = 768, KK = 8;
    const int row = blockIdx.x;

    __shared__ __align__(16) float ks[KK * E];   // 24 KB: gathered rows
    __shared__ float sc[KK];

    const long long* __restrict__ ip = idx + (size_t)row * KK;

#if USE_ASYNC_LDS
    // Each thread issues 6 x B128 async global->LDS transfers.
    // 768 % 4 == 0, so every 16B chunk stays inside one embedding row.
    for (int t4 = threadIdx.x; t4 < (KK * E) / 4; t4 += 256) {
        const int k = t4 / (E / 4);             // 0..7
        const int e = (t4 - k * (E / 4)) * 4;   // 0..764, multiple of 4
        const float* gsrc = w2v + (size_t)ip[k] * E + e;
        __builtin_amdgcn_global_load_async_to_lds_b128(
            (__attribute__((address_space(1))) v4i*)(uintptr_t)gsrc,
            (__attribute__((address_space(3))) v4i*)(uint32_t)(uintptr_t)&ks[t4 * 4],
            0, 0);
    }
#if __has_builtin(__builtin_amdgcn_s_wait_asynccnt)
    __builtin_amdgcn_s_wait_asynccnt(0);
#else
    asm volatile("s_wait_asynccnt 0x0" ::: "memory");
#endif
#else
    for (int t = threadIdx.x; t < KK * E; t += 256) {
        const int k = t / E;
        const int e = t - k * E;
        ks[t] = w2v[(size_t)ip[k] * E + e];
    }
#endif
    __syncthreads();

    const int wave = threadIdx.x >> 5;
    const int lane = threadIdx.x & 31;
    const float* __restrict__ q = Q + (size_t)row * E;

    // Wave w computes score[w] = dot(q, ks[w]) with a wave32 xor-reduction.
    float p = 0.f;
    for (int e = lane; e < E; e += 32)
        p += q[e] * ks[wave * E + e];
    #pragma unroll
    for (int off = 16; off > 0; off >>= 1)
        p += __shfl_xor(p, off, 32);
    if (lane == 0) sc[wave] = p;
    __syncthreads();

    // Softmax over 8 scores (computed redundantly per thread -- cheap).
    float m = sc[0];
    #pragma unroll
    for (int k = 1; k < KK; ++k) m = fmaxf(m, sc[k]);
    float a[KK];
    float s = 0.f;
    #pragma unroll
    for (int k = 0; k < KK; ++k) { a[k] = __expf(sc[k] - m); s += a[k]; }
    const float inv = 1.f / s;

    float* __restrict__ o = out + (size_t)row * E;
    for (int e = threadIdx.x; e < E; e += 256) {
        float acc = 0.f;
        #pragma unroll
        for (int k = 0; k < KK; ++k) acc += a[k] * ks[k * E + e];
        o[e] = acc * inv;
    }
}

extern "C" void kernel_launch(void* const* d_in, const int* in_sizes, int n_in,
                              void* d_out, int out_size, void* d_ws, size_t ws_size,
                              hipStream_t stream) {
    const float*     hidden = (const float*)d_in[0];     // [32,256,768] f32
    const long long* sim    = (const long long*)d_in[1]; // [32,256,8]   i64
    const float*     w2v    = (const float*)d_in[2];     // [50000,768]  f32
    const float*     weight = (const float*)d_in[3];     // [768,768]    f32
    float*           out    = (float*)d_out;             // [32,256,768] f32
    float*           qws    = (float*)d_ws;              // 25.2 MB scratch

    // GEMM: 256 x 24 = 6144 (32x32) tiles, 8 waves/WG -> 768 workgroups.
    qproj_wmma_f32<<<768, 256, 0, stream>>>(hidden, weight, qws);

    // Attention: one WG per (b,l) row.
    attn_softmax_kernel<<<8192, 256, 0, stream>>>(qws, sim, w2v, out);
}